// IndexedLinear_36507222016127
// MI455X (gfx1250) — compile-verified
//
#include <hip/hip_runtime.h>

#define BATCH   8
#define NODES   8192
#define CH      128
#define NEXP    64
#define SLICES  8
#define LDST    132   // padded LDS row stride in floats (132 % 64 == 4 -> conflict-free)

typedef __attribute__((ext_vector_type(2))) float v2f;
typedef __attribute__((ext_vector_type(4))) float v4f;
typedef __attribute__((ext_vector_type(8))) float v8f;

#if defined(__HIP_DEVICE_COMPILE__) && __has_builtin(__builtin_amdgcn_global_load_async_to_lds_b128)
#define HAVE_ASYNC_LD 1
#else
#define HAVE_ASYNC_LD 0
#endif

#if defined(__HIP_DEVICE_COMPILE__) && __has_builtin(__builtin_amdgcn_global_store_async_from_lds_b128)
#define HAVE_ASYNC_ST 1
#else
#define HAVE_ASYNC_ST 0
#endif

#if HAVE_ASYNC_LD || HAVE_ASYNC_ST
typedef int v4i_b128 __attribute__((vector_size(16)));
typedef __attribute__((address_space(1))) v4i_b128* gb128_t;   // global (prints as __device__)
typedef __attribute__((address_space(3))) v4i_b128* lb128_t;   // LDS    (prints as __shared__)
#endif

// Issue one 16-byte global -> LDS copy (async engine if available).
__device__ __forceinline__ void copy_b128(const float* gsrc, float* ldst) {
#if HAVE_ASYNC_LD
    __builtin_amdgcn_global_load_async_to_lds_b128((gb128_t)gsrc, (lb128_t)ldst, 0, 0);
#else
    *(v4f*)ldst = *(const v4f*)gsrc;
#endif
}

// Issue one 16-byte LDS -> global copy (async engine if available).
__device__ __forceinline__ void store_b128(float* gdst, const float* lsrc) {
#if HAVE_ASYNC_ST
    __builtin_amdgcn_global_store_async_from_lds_b128((gb128_t)gdst, (lb128_t)lsrc, 0, 0);
#else
    *(v4f*)gdst = *(const v4f*)lsrc;
#endif
}

// Drain ASYNCcnt (no-op in pure-fallback path) + compiler memory barrier.
__device__ __forceinline__ void copy_fence() {
#if HAVE_ASYNC_LD || HAVE_ASYNC_ST
#if __has_builtin(__builtin_amdgcn_s_wait_asynccnt)
    __builtin_amdgcn_s_wait_asynccnt(0);
#else
    asm volatile("s_wait_asynccnt 0x0" ::: "memory");
#endif
    asm volatile("" ::: "memory");
#endif
}

// d_ws int layout:
//   [0..63]          per-expert counts
//   [64..128]        exclusive prefix offsets (65 entries)
//   [192..255]       scatter cursors
//   [256..256+NODES) node indices sorted by expert

__global__ void init_counts(int* ws) {
    if (threadIdx.x < NEXP) ws[threadIdx.x] = 0;
}

__global__ void count_nodes(const int* __restrict__ sel, int* ws) {
    int n = blockIdx.x * blockDim.x + threadIdx.x;
    if (n < NODES) atomicAdd(&ws[sel[n] & (NEXP - 1)], 1);
}

__global__ void scan_counts(int* ws) {
    if (threadIdx.x == 0) {
        int run = 0;
        for (int i = 0; i < NEXP; ++i) {
            ws[64 + i]  = run;
            ws[192 + i] = run;
            run += ws[i];
        }
        ws[64 + NEXP] = run;
    }
}

__global__ void scatter_nodes(const int* __restrict__ sel, int* ws) {
    int n = blockIdx.x * blockDim.x + threadIdx.x;
    if (n < NODES) {
        int e   = sel[n] & (NEXP - 1);
        int pos = atomicAdd(&ws[192 + e], 1);
        ws[256 + pos] = n;
    }
}

// One block = (expert e, slice s). 8 waves; each wave independently processes
// node-pairs of expert e. Per pair: D(128 x 16) = W[e](128x128) * x^T(128x16),
// 16 columns = {node0,node1} x 8 batches -> full 16x16 WMMA tile utilization.
__global__ __launch_bounds__(256) void moe_gemm(
    const float* __restrict__ x, const float* __restrict__ W,
    const int* __restrict__ ws, float* __restrict__ out) {

    __shared__ float Wl[128 * LDST];      // W[e], row i, padded stride
    __shared__ float xl[8][16 * LDST];    // per-wave x-pair slab, reused as D-staging

    const int e    = blockIdx.x / SLICES;
    const int s    = blockIdx.x % SLICES;
    const int tid  = threadIdx.x;
    const int wave = tid >> 5;
    const int lane = tid & 31;

    // ---- stage W[e] (64 KB) into LDS via async copy engine ----
    const float* Wg = W + (size_t)e * CH * CH;
    #pragma unroll
    for (int k = 0; k < 16; ++k) {
        int g   = (k * 256 + tid) * 4;     // flat float index, never crosses a row
        int row = g >> 7;
        int col = g & 127;
        copy_b128(Wg + g, &Wl[row * LDST + col]);
    }
    copy_fence();
    __syncthreads();

    const int base   = ws[64 + e];
    const int cnt    = ws[64 + e + 1] - base;
    const int npairs = (cnt + 1) >> 1;
    const int* order = ws + 256 + base;

    float* xw = xl[wave];
    const int cL = lane & 15;   // B/D column index for this lane
    const int hL = lane >> 4;   // half-wave K/M group

    for (int p = s * 8 + wave; p < npairs; p += 64) {
        __builtin_prefetch(&order[2 * p + 128], 0, 0);   // global_prefetch_b8

        int  n0   = order[2 * p];
        bool has1 = (2 * p + 1) < cnt;
        int  n1   = has1 ? order[2 * p + 1] : n0;

        const float* x0 = x + (size_t)n0 * CH;
        const float* x1 = x + (size_t)n1 * CH;

        // ---- stage 16 x-columns (8 KB) into this wave's LDS slab (async) ----
        #pragma unroll
        for (int c = 0; c < 16; ++c) {
            const float* src = ((c < 8) ? x0 : x1) + (size_t)(c & 7) * NODES * CH;
            copy_b128(src + lane * 4, &xw[c * LDST + lane * 4]);
        }
        copy_fence();

        // ---- all 32 B-fragments (K = 0..127 step 4) into registers ----
        v2f B[32];
        #pragma unroll
        for (int k = 0; k < 32; ++k)
            B[k] = *(const v2f*)&xw[cL * LDST + k * 4 + 2 * hL];

        // ---- 8 row-tiles of D, 32 K-steps each: 256 WMMAs per pair ----
        for (int t = 0; t < 8; ++t) {
            v8f acc = {0.f, 0.f, 0.f, 0.f, 0.f, 0.f, 0.f, 0.f};
            #pragma unroll
            for (int k = 0; k < 32; ++k) {
                v2f a = *(const v2f*)&Wl[(t * 16 + cL) * LDST + k * 4 + 2 * hL];
                acc = __builtin_amdgcn_wmma_f32_16x16x4_f32(
                        false, a, false, B[k], (short)0, acc, false, false);
            }
            // stash D tile transposed into the slab: column cL, rows i = t*16 + 8*hL + v
            v4f lo = {acc[0], acc[1], acc[2], acc[3]};
            v4f hi = {acc[4], acc[5], acc[6], acc[7]};
            float* d = &xw[cL * LDST + t * 16 + 8 * hL];
            *(v4f*)d       = lo;
            *(v4f*)(d + 4) = hi;
        }

        // ---- coalesced 512B column stores to out[b][n][0..127] (async LDS->global) ----
        #pragma unroll
        for (int c = 0; c < 16; ++c) {
            if (c >= 8 && !has1) break;   // odd pair: second node invalid
            float* dst = out + (size_t)(c & 7) * NODES * CH
                             + (size_t)((c < 8) ? n0 : n1) * CH;
            store_b128(dst + lane * 4, &xw[c * LDST + lane * 4]);
        }
        // drain async stores before the slab is overwritten by the next pair
        copy_fence();
    }
}

extern "C" void kernel_launch(void* const* d_in, const int* in_sizes, int n_in,
                              void* d_out, int out_size, void* d_ws, size_t ws_size,
                              hipStream_t stream) {
    const float* x   = (const float*)d_in[0];
    const int*   sel = (const int*)d_in[1];
    const float* W   = (const float*)d_in[2];
    float*       out = (float*)d_out;
    int*         ws  = (int*)d_ws;

    init_counts  <<<1,            64,  0, stream>>>(ws);
    count_nodes  <<<NODES / 256,  256, 0, stream>>>(sel, ws);
    scan_counts  <<<1,            1,   0, stream>>>(ws);
    scatter_nodes<<<NODES / 256,  256, 0, stream>>>(sel, ws);
    moe_gemm     <<<NEXP * SLICES, 256, 0, stream>>>(x, W, ws, out);
}